// NGCFConv_45715631898813
// MI455X (gfx1250) — compile-verified
//
#include <hip/hip_runtime.h>

typedef __attribute__((ext_vector_type(2))) float v2f;
typedef __attribute__((ext_vector_type(4))) float v4f;
typedef __attribute__((ext_vector_type(8))) float v8f;

#define N_NODES 100000
#define N_EDGES 1600000
#define D 128

// ---------------------------------------------------------------------------
// Kernel 1: zero the SpMM accumulator (ws) — required every call (atomics).
// ---------------------------------------------------------------------------
__global__ void ngcf_zero_f32(float* __restrict__ p, int n) {
    int i = blockIdx.x * blockDim.x + threadIdx.x;
    int stride = gridDim.x * blockDim.x;
    for (; i < n; i += stride) p[i] = 0.0f;
}

// ---------------------------------------------------------------------------
// Kernel 2: COO SpMM scatter. One wave32 per edge; each lane handles 4 floats.
// Guaranteed hardware non-returning f32 atomics at DEVICE scope via inline asm
// (WGP-scope default would RMW in the non-coherent near cache; non-returning
// rides STOREcnt; s_endpgm's implicit wait-idle covers completion).
// Edge index/value streams are non-temporal so the 19 MB of COO data does not
// evict the L2-resident prev (51 MB) + L (51 MB) working set (L2 = 192 MB).
// ---------------------------------------------------------------------------
__global__ __launch_bounds__(256) void ngcf_spmm_scatter(
    const int* __restrict__ rows, const int* __restrict__ cols,
    const float* __restrict__ vals, const float* __restrict__ prev,
    float* __restrict__ L) {
    int e = blockIdx.x * 8 + (threadIdx.x >> 5);
    if (e >= N_EDGES) return;
    int lane = threadIdx.x & 31;

    int   r = __builtin_nontemporal_load(rows + e);
    int   c = __builtin_nontemporal_load(cols + e);
    float v = __builtin_nontemporal_load(vals + e);

    const v4f* src = (const v4f*)(prev + (size_t)c * D);
    v4f m = src[lane] * v;

    float* dst = L + (size_t)r * D + lane * 4;
    asm volatile(
        "global_atomic_add_f32 %0, %1, off scope:SCOPE_DEV\n\t"
        "global_atomic_add_f32 %0, %2, off offset:4 scope:SCOPE_DEV\n\t"
        "global_atomic_add_f32 %0, %3, off offset:8 scope:SCOPE_DEV\n\t"
        "global_atomic_add_f32 %0, %4, off offset:12 scope:SCOPE_DEV"
        :: "v"(dst), "v"(m.x), "v"(m.y), "v"(m.z), "v"(m.w)
        : "memory");
}

// ---------------------------------------------------------------------------
// Kernel 3: fused dense part with fp32 WMMA.
//   out = L @ W1 + b1 + (prev*L) @ W2 + b2
// Treated as a single K=256 GEMM: [Hl | Hr] @ [W1 ; W2].
// Block = 64 threads (2 waves). Each wave owns one 16-row tile.
// 100000 rows = 6250 tiles = 3125 blocks * 2 waves exactly (no divergence;
// EXEC is all-1s for every WMMA as the ISA requires).
//
// LDS staging: sA[wave][16 rows][260], stride 260 (mod 64 = 4) makes the
// A-fragment pattern conflict-free across all 64 banks with 8B ds_load_b64.
// L/prev are read exactly once per element -> non-temporal loads; out is
// written once -> non-temporal stores. W1/W2/b1/b2 are reused by every block
// and stay regular-temporal (they live in WGP$/L2).
// ---------------------------------------------------------------------------
#define SROW 260
#define WPB  2

__global__ __launch_bounds__(64) void ngcf_wmma_gemm(
    const float* __restrict__ L, const float* __restrict__ prev,
    const float* __restrict__ W1, const float* __restrict__ b1,
    const float* __restrict__ W2, const float* __restrict__ b2,
    float* __restrict__ out) {
    __shared__ float sA[WPB][16][SROW];

    const int wave = threadIdx.x >> 5;
    const int lane = threadIdx.x & 31;
    const int tile = blockIdx.x * WPB + wave;   // 16-row tile index, < 6250
    const int m    = lane & 15;                  // row within tile (A) / col (B,C)
    const int hi   = lane >> 4;                  // half-wave selector

    // ---- Stage Hl (cols 0..127) and Hr = prev*L (cols 128..255) in LDS ----
    const float* Lt = L    + (size_t)tile * 16 * D;
    const float* Pt = prev + (size_t)tile * 16 * D;
    for (int r = 0; r < 16; ++r) {
        v4f l = __builtin_nontemporal_load((const v4f*)(Lt + r * D) + lane);
        v4f p = __builtin_nontemporal_load((const v4f*)(Pt + r * D) + lane);
        v4f h = l * p;
        int k = lane * 4;
        sA[wave][r][k + 0] = l.x;
        sA[wave][r][k + 1] = l.y;
        sA[wave][r][k + 2] = l.z;
        sA[wave][r][k + 3] = l.w;
        sA[wave][r][128 + k + 0] = h.x;
        sA[wave][r][128 + k + 1] = h.y;
        sA[wave][r][128 + k + 2] = h.z;
        sA[wave][r][128 + k + 3] = h.w;
    }
    __syncthreads();   // all waves always active -> safe

    const float* sa = &sA[wave][m][0];

    // ---- 8 N-tiles of 16 columns each ----
    for (int j = 0; j < 8; ++j) {
        const int n = j * 16 + m;               // global output column (for B/C/D)
        v8f c = {};

        // K = 0..127 : Hl @ W1
        const float* W1n = W1 + n;
        #pragma unroll 8
        for (int kk = 0; kk < 32; ++kk) {
            const int k0 = kk * 4 + hi * 2;     // A/B fragment K base per half-wave
            v2f a = *(const v2f*)(sa + k0);
            v2f b;
            b.x = W1n[(k0 + 0) * D];
            b.y = W1n[(k0 + 1) * D];
            c = __builtin_amdgcn_wmma_f32_16x16x4_f32(
                    false, a, false, b, (short)0, c, false, false);
        }

        // K = 128..255 : Hr @ W2
        const float* W2n = W2 + n;
        #pragma unroll 8
        for (int kk = 0; kk < 32; ++kk) {
            const int k0 = kk * 4 + hi * 2;
            v2f a = *(const v2f*)(sa + 128 + k0);
            v2f b;
            b.x = W2n[(k0 + 0) * D];
            b.y = W2n[(k0 + 1) * D];
            c = __builtin_amdgcn_wmma_f32_16x16x4_f32(
                    false, a, false, b, (short)0, c, false, false);
        }

        // ---- Store: C/D layout = 8 VGPRs; lane<16 rows 0..7, lane>=16 rows 8..15
        const float bias = b1[n] + b2[n];
        float* ocol = out + (size_t)tile * 16 * D + n;
        #pragma unroll
        for (int v = 0; v < 8; ++v) {
            int mrow = v + hi * 8;
            __builtin_nontemporal_store(c[v] + bias, ocol + (size_t)mrow * D);
        }
    }
}

// ---------------------------------------------------------------------------
// Host-side launcher (graph-capture safe: kernels only, all on `stream`).
// Inputs (dict order): edge_rows, edge_cols, edge_vals, prev_embeddings,
//                      W1, b1, W2, b2
// ---------------------------------------------------------------------------
extern "C" void kernel_launch(void* const* d_in, const int* in_sizes, int n_in,
                              void* d_out, int out_size, void* d_ws, size_t ws_size,
                              hipStream_t stream) {
    const int*   erows = (const int*)d_in[0];
    const int*   ecols = (const int*)d_in[1];
    const float* evals = (const float*)d_in[2];
    const float* prev  = (const float*)d_in[3];
    const float* W1    = (const float*)d_in[4];
    const float* b1    = (const float*)d_in[5];
    const float* W2    = (const float*)d_in[6];
    const float* b2    = (const float*)d_in[7];
    float*       out   = (float*)d_out;
    float*       L     = (float*)d_ws;      // [N_NODES * D] accumulator

    (void)in_sizes; (void)n_in; (void)out_size; (void)ws_size;

    // 1) zero accumulator
    ngcf_zero_f32<<<2048, 256, 0, stream>>>(L, N_NODES * D);

    // 2) SpMM scatter: wave per edge, hardware f32 atomics at device scope
    ngcf_spmm_scatter<<<N_EDGES / 8, 256, 0, stream>>>(erows, ecols, evals, prev, L);

    // 3) fused fp32-WMMA GEMM + biases
    ngcf_wmma_gemm<<<(N_NODES / 16) / WPB, 64, 0, stream>>>(L, prev, W1, b1, W2, b2, out);
}